// S4_Model_16535624089956
// MI455X (gfx1250) — compile-verified
//
#include <hip/hip_runtime.h>
#include <math.h>

// ---------------------------------------------------------------------------
// S4 model forward for MI455X (gfx1250, wave32).
//   encoder GEMM  (WMMA f32 16x16x4, LDS-staged via async global->LDS copies)
//   4x [ S4 diagonal-SSM recurrence (register-resident VALU) + gelu
//        -> GLU projection GEMM (WMMA, sigmoid gate fused in epilogue) ]
//   decoder GEMM
// ---------------------------------------------------------------------------

#define BATCH   32
#define LSEQ    4096
#define DIN     128
#define HMODEL  256
#define DOUT    128
#define NLAYERS 4
#define NSTATE  32

#define BLK_M   128          // rows per block (8 waves x 16)
#define BLK_N   64           // output cols per block (4 wmma tiles)
#define PAD     4            // LDS row padding (floats) -> conflict-free frags

typedef float v2f __attribute__((ext_vector_type(2)));
typedef float v4f __attribute__((ext_vector_type(4)));
typedef float v8f __attribute__((ext_vector_type(8)));

__device__ __forceinline__ float gelu_tanh(float x) {
    float inner = 0.7978845608028654f * (x + 0.044715f * x * x * x);
    return 0.5f * x * (1.0f + tanhf(inner));
}
__device__ __forceinline__ float sigmoidf(float x) {
    return 1.0f / (1.0f + expf(-x));
}

// Async 16-byte global -> LDS copy (CDNA5, ASYNCcnt-tracked).
// the LDS destination address comes from the lds_off VGPR; generic shared pointers carry the LDS
// byte offset in their low 32 bits on AMDGPU.
__device__ __forceinline__ void async_g2l_b128(uint32_t lds_off, const float* gaddr) {
    asm volatile("global_load_async_to_lds_b128 %0, %1, off"
                 :: "v"(lds_off), "v"(gaddr)
                 : "memory");
}
__device__ __forceinline__ void wait_asynccnt0() {
    asm volatile("s_wait_asynccnt 0" ::: "memory");
}

// ---------------------------------------------------------------------------
// Tiled GEMM:  Out[M, NOUT] = act( A[M, K] @ W[rows, K]^T + bias )
//   plain mode: rows == NOUT, act = identity (+bias)
//   GLU mode  : rows == 2*NOUT, out = (h1+b1) * sigmoid(h2+b2)
// Block: 256 threads = 8 waves; block tile 128 x 64.  A tile + B panel(s) are
// staged in LDS with async b128 copies; each wave runs V_WMMA_F32_16X16X4_F32
// on ds_load_b64 fragments (bank-conflict-free via +4 row padding).
// ---------------------------------------------------------------------------
template <int K, bool GLU>
__global__ __launch_bounds__(256) void gemm_wmma_kernel(
    const float* __restrict__ A,     // [M, K]
    const float* __restrict__ W,     // [(GLU?2:1)*NOUT, K]
    const float* __restrict__ bias,  // [(GLU?2:1)*NOUT]
    float* __restrict__ Out,         // [M, NOUT]
    int M, int NOUT)
{
    constexpr int LDK = K + PAD;     // padded LDS row stride (floats)
    extern __shared__ float smem[];
    float* sA  = smem;                       // [BLK_M][LDK]
    float* sB  = sA + BLK_M * LDK;           // [BLK_N][LDK]
    float* sB2 = sB + BLK_N * LDK;           // [BLK_N][LDK] (GLU only)

    const int tid = threadIdx.x;
    const int nblkN = NOUT / BLK_N;
    const int gm = blockIdx.x / nblkN;       // M block
    const int gn = blockIdx.x % nblkN;       // N block

    // ---- stage A tile: BLK_M rows x K --------------------------------------
    for (int i = tid; i < BLK_M * (K / 4); i += 256) {
        const int row = i / (K / 4);
        const int kc  = (i % (K / 4)) * 4;
        async_g2l_b128((uint32_t)(uintptr_t)&sA[row * LDK + kc],
                       A + (size_t)(gm * BLK_M + row) * K + kc);
    }
    // ---- stage B panel(s): BLK_N weight rows x K ---------------------------
    for (int i = tid; i < BLK_N * (K / 4); i += 256) {
        const int c  = i / (K / 4);
        const int kc = (i % (K / 4)) * 4;
        async_g2l_b128((uint32_t)(uintptr_t)&sB[c * LDK + kc],
                       W + (size_t)(gn * BLK_N + c) * K + kc);
        if constexpr (GLU) {
            async_g2l_b128((uint32_t)(uintptr_t)&sB2[c * LDK + kc],
                           W + (size_t)(gn * BLK_N + c + NOUT) * K + kc);
        }
    }
    wait_asynccnt0();
    __syncthreads();

    // ---- WMMA compute ------------------------------------------------------
    const int wv   = tid >> 5;               // wave -> 16-row slice
    const int lane = tid & 31;
    const int r    = lane & 15;              // A row / B col within tile
    const int kh   = (lane >> 4) * 2;        // K sub-offset: 0 or 2

    const float* aRow = &sA[(wv * 16 + r) * LDK + kh];

    v8f acc[4] = {};
    v8f acc2[GLU ? 4 : 1] = {};

#pragma unroll 8
    for (int k = 0; k < K; k += 4) {
        v2f a;
        a.x = aRow[k];
        a.y = aRow[k + 1];
#pragma unroll
        for (int t = 0; t < 4; ++t) {
            const float* bp = &sB[(t * 16 + r) * LDK + kh + k];
            v2f b; b.x = bp[0]; b.y = bp[1];
            acc[t] = __builtin_amdgcn_wmma_f32_16x16x4_f32(
                false, a, false, b, (short)0, acc[t], false, false);
            if constexpr (GLU) {
                const float* bp2 = &sB2[(t * 16 + r) * LDK + kh + k];
                v2f b2; b2.x = bp2[0]; b2.y = bp2[1];
                acc2[t] = __builtin_amdgcn_wmma_f32_16x16x4_f32(
                    false, a, false, b2, (short)0, acc2[t], false, false);
            }
        }
    }

    // ---- epilogue ----------------------------------------------------------
    // D layout: VGPR j, lanes 0-15 -> M=j ; lanes 16-31 -> M=j+8 ; N = lane&15
    const int rowbase = gm * BLK_M + wv * 16 + (lane >> 4) * 8;
#pragma unroll
    for (int t = 0; t < 4; ++t) {
        const int col = gn * BLK_N + t * 16 + r;
        const float b1 = bias[col];
        const float b2 = GLU ? bias[col + NOUT] : 0.0f;
#pragma unroll
        for (int j = 0; j < 8; ++j) {
            float v;
            if constexpr (GLU) {
                float h1 = acc[t][j] + b1;
                float h2 = acc2[t][j] + b2;
                v = h1 * sigmoidf(h2);
            } else {
                v = acc[t][j] + b1;
            }
            Out[(size_t)(rowbase + j) * NOUT + col] = v;
        }
    }
}

// ---------------------------------------------------------------------------
// S4 layer core: recurrent (ZOH-discretized) diagonal SSM, equivalent to the
// reference's FFT convolution.  One thread per (b, h); 32 complex modes in
// registers; lanes span consecutive h -> coalesced activation streaming.
// ---------------------------------------------------------------------------
__global__ __launch_bounds__(256) void s4_scan_kernel(
    const float* __restrict__ z,        // [B*L, H]
    const float* __restrict__ log_dt,   // [H]
    const float* __restrict__ A_log_re, // [H, N]
    const float* __restrict__ A_im,     // [H, N]
    const float* __restrict__ C_re,     // [H, N]
    const float* __restrict__ C_im,     // [H, N]
    const float* __restrict__ Dp,       // [H]
    float* __restrict__ y)              // [B*L, H]
{
    const int tid = blockIdx.x * blockDim.x + threadIdx.x;
    if (tid >= BATCH * HMODEL) return;
    const int b = tid >> 8;
    const int h = tid & (HMODEL - 1);

    const float dt  = expf(log_dt[h]);
    const float dpv = Dp[h];

    float wre[NSTATE], wim[NSTATE], ccr[NSTATE], cci[NSTATE];
    float sre[NSTATE], sim[NSTATE];
#pragma unroll
    for (int n = 0; n < NSTATE; ++n) {
        const int idx = h * NSTATE + n;
        const float Are = -expf(A_log_re[idx]);
        const float Aim = A_im[idx];
        const float mag = expf(dt * Are);
        const float ang = dt * Aim;
        const float wr  = mag * cosf(ang);
        const float wi  = mag * sinf(ang);
        wre[n] = wr; wim[n] = wi;
        const float inv = 1.0f / (Are * Are + Aim * Aim);
        const float nr  = wr - 1.0f, ni = wi;
        const float qr  = (nr * Are + ni * Aim) * inv;
        const float qi  = (ni * Are - nr * Aim) * inv;
        const float cr  = C_re[idx], ci = C_im[idx];
        ccr[n] = cr * qr - ci * qi;
        cci[n] = cr * qi + ci * qr;
        sre[n] = 0.0f; sim[n] = 0.0f;
    }

    const float* zp = z + (size_t)b * LSEQ * HMODEL + h;
    float*       yp = y + (size_t)b * LSEQ * HMODEL + h;

    for (int l = 0; l < LSEQ; ++l) {
        const float u = zp[(size_t)l * HMODEL];
        float acc = 0.0f;
#pragma unroll
        for (int n = 0; n < NSTATE; ++n) {
            const float nsr = fmaf(wre[n], sre[n], -wim[n] * sim[n]) + u;
            const float nsi = fmaf(wre[n], sim[n],  wim[n] * sre[n]);
            sre[n] = nsr; sim[n] = nsi;
            acc = fmaf(ccr[n], nsr, acc);
            acc = fmaf(-cci[n], nsi, acc);
        }
        const float v = 2.0f * acc + u * dpv;
        yp[(size_t)l * HMODEL] = gelu_tanh(v);
    }
}

// ---------------------------------------------------------------------------
extern "C" void kernel_launch(void* const* d_in, const int* in_sizes, int n_in,
                              void* d_out, int out_size, void* d_ws, size_t ws_size,
                              hipStream_t stream)
{
    const float* x        = (const float*)d_in[0];
    const float* enc_W    = (const float*)d_in[1];
    const float* enc_b    = (const float*)d_in[2];
    const float* log_dt   = (const float*)d_in[3];
    const float* A_log_re = (const float*)d_in[4];
    const float* A_im     = (const float*)d_in[5];
    const float* C_re     = (const float*)d_in[6];
    const float* C_im     = (const float*)d_in[7];
    const float* Dp       = (const float*)d_in[8];
    const float* W_out    = (const float*)d_in[9];
    const float* b_out    = (const float*)d_in[10];
    const float* dec_W    = (const float*)d_in[11];
    const float* dec_b    = (const float*)d_in[12];
    float* out = (float*)d_out;

    const int M = BATCH * LSEQ;                 // 131072 rows
    float* zbuf = (float*)d_ws;                 // [M, 256]
    float* ybuf = zbuf + (size_t)M * HMODEL;    // [M, 256]

    const dim3 blk(256);

    // LDS bytes: (BLK_M + (GLU?2:1)*BLK_N) * (K+PAD) * 4
    const size_t lds_enc = (size_t)(BLK_M + BLK_N)     * (DIN    + PAD) * 4; //  ~99 KB
    const size_t lds_glu = (size_t)(BLK_M + 2 * BLK_N) * (HMODEL + PAD) * 4; // ~260 KB
    const size_t lds_dec = (size_t)(BLK_M + BLK_N)     * (HMODEL + PAD) * 4; // ~195 KB

    // Encoder: z = x @ enc_W.T + enc_b        (K=128, NOUT=256)
    gemm_wmma_kernel<DIN, false>
        <<<(M / BLK_M) * (HMODEL / BLK_N), blk, lds_enc, stream>>>(
            x, enc_W, enc_b, zbuf, M, HMODEL);

    for (int layer = 0; layer < NLAYERS; ++layer) {
        // S4 recurrence + skip + gelu : zbuf -> ybuf
        s4_scan_kernel<<<(BATCH * HMODEL) / 256, blk, 0, stream>>>(
            zbuf,
            log_dt   + (size_t)layer * HMODEL,
            A_log_re + (size_t)layer * HMODEL * NSTATE,
            A_im     + (size_t)layer * HMODEL * NSTATE,
            C_re     + (size_t)layer * HMODEL * NSTATE,
            C_im     + (size_t)layer * HMODEL * NSTATE,
            Dp       + (size_t)layer * HMODEL,
            ybuf);

        // GLU projection: z = GLU(y @ W_out.T + b_out)   (K=256, NOUT=256)
        gemm_wmma_kernel<HMODEL, true>
            <<<(M / BLK_M) * (HMODEL / BLK_N), blk, lds_glu, stream>>>(
                ybuf,
                W_out + (size_t)layer * 2 * HMODEL * HMODEL,
                b_out + (size_t)layer * 2 * HMODEL,
                zbuf, M, HMODEL);
    }

    // Decoder: out = z @ dec_W.T + dec_b     (K=256, NOUT=128)
    gemm_wmma_kernel<HMODEL, false>
        <<<(M / BLK_M) * (DOUT / BLK_N), blk, lds_dec, stream>>>(
            zbuf, dec_W, dec_b, out, M, DOUT);
}